// YoloLoss_39024072851595
// MI455X (gfx1250) — compile-verified
//
#include <hip/hip_runtime.h>
#include <math.h>

// ---------------------------------------------------------------------------
// YOLO v1 loss on MI455X (gfx1250, wave32).
// Memory-bound: ~176 MB streamed once -> ~7.6us floor @ 23.3 TB/s.
// Strategy:
//   * per-block contiguous chunks staged global->LDS with
//     GLOBAL_LOAD_ASYNC_TO_LDS_B128 (ASYNCcnt) -> perfectly coalesced VMEM
//   * per-cell math out of LDS (stride-30 floats => conflict-free banking)
//   * wave32 shfl_xor reduction, cross-wave reduction via
//     V_WMMA_F32_16X16X4_F32 (A = ones => fp32 column sums)
//   * deterministic two-stage global reduction through d_ws
// ---------------------------------------------------------------------------

static constexpr int   kN     = 16384;
static constexpr int   kS     = 7;
static constexpr int   kCells = kN * kS * kS;        // 802816
static constexpr int   kBlock = 256;                 // 8 wave32 waves
static constexpr int   kGrid  = kCells / kBlock;     // 3136 (exact)
static constexpr float kEps   = 1e-6f;

static constexpr int kPredF4 = kBlock * 30 / 4;      // 1920 float4 per block
static constexpr int kTgtF4  = kBlock * 25 / 4;      // 1600 float4 per block

typedef float v2f __attribute__((ext_vector_type(2)));
typedef float v4f __attribute__((ext_vector_type(4)));
typedef float v8f __attribute__((ext_vector_type(8)));
typedef int   v4i __attribute__((ext_vector_type(4)));

#define GLOBAL_AS __attribute__((address_space(1)))
#define LDS_AS    __attribute__((address_space(3)))

#define HAS_WMMA_F32X4 __has_builtin(__builtin_amdgcn_wmma_f32_16x16x4_f32)
#define HAS_ASYNC_LDS  __has_builtin(__builtin_amdgcn_global_load_async_to_lds_b128)

__device__ __forceinline__ void wait_async_and_barrier() {
#if HAS_ASYNC_LDS
#if __has_builtin(__builtin_amdgcn_s_wait_asynccnt)
    __builtin_amdgcn_s_wait_asynccnt(0);
#else
    asm volatile("s_wait_asynccnt 0x0" ::: "memory");
#endif
#endif
    __syncthreads();
}

__device__ __forceinline__ float wave_sum(float v) {
    v += __shfl_xor(v, 16, 32);
    v += __shfl_xor(v,  8, 32);
    v += __shfl_xor(v,  4, 32);
    v += __shfl_xor(v,  2, 32);
    v += __shfl_xor(v,  1, 32);
    return v;
}

// Cross-wave (8 waves x 5 accumulators) reduction with V_WMMA_F32_16X16X4_F32.
// P: 8x16 LDS tile, row = wave, col = accumulator index (cols 5..15 are 0).
// With A == ones(16x4): D[m][n] = sum_k B[k][n] (column sum); two WMMAs chained
// via C cover all 8 rows. Invariant to K-permutation of the B register layout.
// Calling wave must be fully active (EXEC all ones).
__device__ __forceinline__ float wmma_colsum8(const float (*P)[16], int lane) {
#if HAS_WMMA_F32X4
    const int half = lane >> 4;
    const int n    = lane & 15;
    v2f a;  a[0] = 1.0f;                 a[1] = 1.0f;
    v2f b0; b0[0] = P[2 * half + 0][n];  b0[1] = P[2 * half + 1][n]; // waves 0..3
    v2f b1; b1[0] = P[2 * half + 4][n];  b1[1] = P[2 * half + 5][n]; // waves 4..7
    v8f c = {0.f, 0.f, 0.f, 0.f, 0.f, 0.f, 0.f, 0.f};
    v8f d = __builtin_amdgcn_wmma_f32_16x16x4_f32(false, a, false, b0,
                                                  (short)0, c, false, false);
    d     = __builtin_amdgcn_wmma_f32_16x16x4_f32(false, a, false, b1,
                                                  (short)0, d, false, false);
    return d[0];
#else
    const int n = lane & 15;
    float s = 0.0f;
    #pragma unroll
    for (int w = 0; w < 8; ++w) s += P[w][n];
    return s;
#endif
}

__device__ __forceinline__ float iou_corner(float ax1, float ay1, float ax2, float ay2,
                                            float bx1, float by1, float bx2, float by2) {
    const float xi1 = fmaxf(ax1, bx1);
    const float yi1 = fmaxf(ay1, by1);
    const float xi2 = fminf(ax2, bx2);
    const float yi2 = fminf(ay2, by2);
    const float inter = fmaxf(xi2 - xi1, 0.0f) * fmaxf(yi2 - yi1, 0.0f);
    const float a1 = (ax2 - ax1) * (ay2 - ay1);
    const float a2 = (bx2 - bx1) * (by2 - by1);
    return inter / (a1 + a2 - inter + kEps);
}

// ---------------------------------------------------------------------------
// Stage 1: per-cell loss terms -> per-block partial sums (5 values / block).
// ---------------------------------------------------------------------------
template <bool ATOMIC>
__global__ void __launch_bounds__(kBlock)
yolo_partial_kernel(const float* __restrict__ pred,
                    const float* __restrict__ tgt,
                    float* __restrict__ partials) {
    __shared__ v4f lds_pred4[kPredF4];     // 30720 B: 256 cells x 30 floats
    __shared__ v4f lds_tgt4[kTgtF4];       // 25600 B: 256 cells x 25 floats
    __shared__ float P[8][16];

    const int tid = threadIdx.x;

    // ---- stage this block's contiguous chunks into LDS (coalesced b128) ----
    const v4f* pg4 = (const v4f*)pred + (size_t)blockIdx.x * kPredF4;
    const v4f* tg4 = (const v4f*)tgt  + (size_t)blockIdx.x * kTgtF4;
#if HAS_ASYNC_LDS
    for (int i = tid; i < kPredF4; i += kBlock)
        __builtin_amdgcn_global_load_async_to_lds_b128(
            (GLOBAL_AS v4i*)(pg4 + i), (LDS_AS v4i*)&lds_pred4[i], 0, 0);
    for (int i = tid; i < kTgtF4; i += kBlock)
        __builtin_amdgcn_global_load_async_to_lds_b128(
            (GLOBAL_AS v4i*)(tg4 + i), (LDS_AS v4i*)&lds_tgt4[i], 0, 0);
#else
    for (int i = tid; i < kPredF4; i += kBlock)
        lds_pred4[i] = __builtin_nontemporal_load(pg4 + i);
    for (int i = tid; i < kTgtF4; i += kBlock)
        lds_tgt4[i] = __builtin_nontemporal_load(tg4 + i);
#endif
    if (tid < 128) ((float*)P)[tid] = 0.0f;
    wait_async_and_barrier();

    // ---- per-cell loss terms (reads from LDS; 32 lanes -> 32 banks) ----
    const int cell = blockIdx.x * kBlock + tid;
    const int rem  = cell % (kS * kS);
    const float col   = (float)(rem % kS);
    const float row   = (float)(rem / kS);
    const float inv_s = 1.0f / (float)kS;

    const float* pv = (const float*)lds_pred4 + tid * 30;
    const float* tv = (const float*)lds_tgt4  + tid * 25;

    // target absolute corner box (scaled by conf)
    const float conf = tv[4];
    const float txc = (tv[0] + col) * inv_s;
    const float tyc = (tv[1] + row) * inv_s;
    const float thw = tv[2] * 0.5f;
    const float thh = tv[3] * 0.5f;
    const float t_x1 = (txc - thw) * conf;
    const float t_y1 = (tyc - thh) * conf;
    const float t_x2 = (txc + thw) * conf;
    const float t_y2 = (tyc + thh) * conf;

    // predicted boxes 1 and 2, IoU vs target
    const float p1xc = (pv[0] + col) * inv_s;
    const float p1yc = (pv[1] + row) * inv_s;
    const float p1hw = pv[2] * 0.5f;
    const float p1hh = pv[3] * 0.5f;
    const float iou1 = iou_corner(t_x1, t_y1, t_x2, t_y2,
                                  p1xc - p1hw, p1yc - p1hh, p1xc + p1hw, p1yc + p1hh);

    const float p2xc = (pv[5] + col) * inv_s;
    const float p2yc = (pv[6] + row) * inv_s;
    const float p2hw = pv[7] * 0.5f;
    const float p2hh = pv[8] * 0.5f;
    const float iou2 = iou_corner(t_x1, t_y1, t_x2, t_y2,
                                  p2xc - p2hw, p2yc - p2hh, p2xc + p2hw, p2yc + p2hh);

    const bool pick2 = iou1 > iou2;
    const float xh = pick2 ? pv[5] : pv[0];
    const float yh = pick2 ? pv[6] : pv[1];
    const float wh = pick2 ? pv[7] : pv[2];
    const float hh = pick2 ? pv[8] : pv[3];
    const float ch = pick2 ? pv[9] : pv[4];

    const bool noobj = (conf == 0.0f);

    const float dxy0 = tv[0] - (noobj ? 0.0f : fmaxf(xh, 0.0f));
    const float dxy1 = tv[1] - (noobj ? 0.0f : fmaxf(yh, 0.0f));
    float s_xy = dxy0 * dxy0 + dxy1 * dxy1;

    const float dwh0 = tv[2] - (noobj ? 0.0f : fmaxf(wh, 0.0f));
    const float dwh1 = tv[3] - (noobj ? 0.0f : fmaxf(hh, 0.0f));
    float s_wh = dwh0 * dwh0 + dwh1 * dwh1;   // sqrt applied after global sum

    const float chc = fminf(fmaxf(ch, 0.0f), 1.0f);
    const float dco = conf - (noobj ? 0.0f : chc);
    const float dcn = conf - (noobj ? chc : 0.0f);
    float s_co = dco * dco;
    float s_cn = dcn * dcn;

    float s_cl = 0.0f;
    #pragma unroll
    for (int j = 0; j < 20; ++j) {
        const float d = tv[5 + j] - (noobj ? 0.0f : pv[10 + j]);
        s_cl += d * d;
    }

    // ---- wave32 butterfly reduction ----
    s_xy = wave_sum(s_xy);
    s_wh = wave_sum(s_wh);
    s_co = wave_sum(s_co);
    s_cn = wave_sum(s_cn);
    s_cl = wave_sum(s_cl);

    // ---- cross-wave reduction via WMMA column-sum ----
    const int wv   = tid >> 5;
    const int lane = tid & 31;
    if (lane == 0) {
        P[wv][0] = s_xy; P[wv][1] = s_wh; P[wv][2] = s_co;
        P[wv][3] = s_cn; P[wv][4] = s_cl;
    }
    __syncthreads();
    if (wv == 0) {                               // whole wave -> EXEC all ones
        const float colsum = wmma_colsum8(P, lane);
        if (lane < 5) {
            if (ATOMIC) atomicAdd(&partials[lane], colsum);
            else        partials[(size_t)blockIdx.x * 5 + lane] = colsum;
        }
    }
}

// ---------------------------------------------------------------------------
// Stage 2: deterministic reduction of kGrid per-block partials + epilogue.
// ---------------------------------------------------------------------------
__global__ void __launch_bounds__(kBlock)
yolo_final_kernel(const float* __restrict__ partials, float* __restrict__ out) {
    float acc[5] = {0.f, 0.f, 0.f, 0.f, 0.f};
    for (int b = threadIdx.x; b < kGrid; b += kBlock) {
        #pragma unroll
        for (int j = 0; j < 5; ++j) acc[j] += partials[(size_t)b * 5 + j];
    }
    #pragma unroll
    for (int j = 0; j < 5; ++j) acc[j] = wave_sum(acc[j]);

    __shared__ float P[8][16];
    __shared__ float T[8];
    if (threadIdx.x < 128) ((float*)P)[threadIdx.x] = 0.0f;
    __syncthreads();
    const int wv   = threadIdx.x >> 5;
    const int lane = threadIdx.x & 31;
    if (lane == 0) {
        #pragma unroll
        for (int j = 0; j < 5; ++j) P[wv][j] = acc[j];
    }
    __syncthreads();
    if (wv == 0) {
        const float colsum = wmma_colsum8(P, lane);
        if (lane < 8) T[lane] = colsum;
    }
    __syncthreads();
    if (threadIdx.x == 0) {
        const float invn = 1.0f / (float)kN;
        const float lxy = T[0] * invn;
        const float lwh = sqrtf(T[1] + kEps) * invn;
        const float lco = T[2] * invn;
        const float lcn = T[3] * invn;
        const float lcl = T[4] * invn;
        out[0] = lxy; out[1] = lwh; out[2] = lco; out[3] = lcn; out[4] = lcl;
        out[5] = 5.0f * lxy + 5.0f * lwh + lco + 0.5f * lcn + lcl;
    }
}

// Fallback epilogue when workspace is too small for per-block partials.
__global__ void yolo_init5_kernel(float* __restrict__ ws) {
    if (threadIdx.x < 5) ws[threadIdx.x] = 0.0f;
}
__global__ void yolo_final5_kernel(const float* __restrict__ ws, float* __restrict__ out) {
    if (threadIdx.x == 0) {
        const float invn = 1.0f / (float)kN;
        const float lxy = ws[0] * invn;
        const float lwh = sqrtf(ws[1] + kEps) * invn;
        const float lco = ws[2] * invn;
        const float lcn = ws[3] * invn;
        const float lcl = ws[4] * invn;
        out[0] = lxy; out[1] = lwh; out[2] = lco; out[3] = lcn; out[4] = lcl;
        out[5] = 5.0f * lxy + 5.0f * lwh + lco + 0.5f * lcn + lcl;
    }
}

extern "C" void kernel_launch(void* const* d_in, const int* in_sizes, int n_in,
                              void* d_out, int out_size, void* d_ws, size_t ws_size,
                              hipStream_t stream) {
    (void)in_sizes; (void)n_in; (void)out_size;
    const float* pred = (const float*)d_in[0];
    const float* tgt  = (const float*)d_in[1];
    float* ws  = (float*)d_ws;
    float* out = (float*)d_out;

    if (ws_size >= (size_t)kGrid * 5 * sizeof(float)) {
        // Deterministic two-stage reduction (fixed summation order per launch).
        yolo_partial_kernel<false><<<kGrid, kBlock, 0, stream>>>(pred, tgt, ws);
        yolo_final_kernel<<<1, kBlock, 0, stream>>>(ws, out);
    } else {
        // Fallback: atomic accumulation into 5 scalars.
        yolo_init5_kernel<<<1, 32, 0, stream>>>(ws);
        yolo_partial_kernel<true><<<kGrid, kBlock, 0, stream>>>(pred, tgt, ws);
        yolo_final5_kernel<<<1, 32, 0, stream>>>(ws, out);
    }
}